// NodeProduct_40819369181354
// MI455X (gfx1250) — compile-verified
//
#include <hip/hip_runtime.h>
#include <stdint.h>

// Problem constants (from reference: B=2, N=768, D_IN=256, D_EDGE=128)
#define B_    2
#define N_    768
#define DIN   256
#define DE    128
#define PAD   4           // LDS row padding (floats) to break 16-row bank conflicts
#define LNEPS 1e-5f

typedef float v2f __attribute__((ext_vector_type(2)));
typedef float v4f __attribute__((ext_vector_type(4)));
typedef float v8f __attribute__((ext_vector_type(8)));

// ---------------------------------------------------------------------------
// One "side" pass: hidden = mask * (xs @ W + bias); Ldst = hidden @ We
// xs: 16 x DIN normalized rows in LDS. hs: 16 x DIN scratch in LDS.
// Uses V_WMMA_F32_16X16X4_F32 tiles (wave32, 4-wave workgroup).
// A fragment (16x4 f32):  lane l -> row = l&15, K pair = 2*(l>>4)
// B fragment (4x16 f32):  lane l -> col = l&15, K row  = 2*(l>>4)
// D tile (16x16 f32):     vgpr v, lane l -> row = v + 8*(l>>4), col = l&15
// ---------------------------------------------------------------------------
__device__ __forceinline__ void gemm_side(
    const float (*xs)[DIN + PAD],
    float (*hs)[DIN + PAD],
    const float* __restrict__ W,     // DIN x DIN  row-major
    const float* __restrict__ bias,  // DIN
    const float* __restrict__ We,    // DIN x DE   row-major
    const float* maskf,              // 16 per-row node masks (0/1)
    float* __restrict__ Ldst,        // out base for this tile, row stride DE
    int wave, int lane)
{
    const int row = lane & 15;            // A-matrix row
    const int kh  = (lane >> 4) << 1;     // K sub-offset: 0 or 2
    const int cl  = lane & 15;            // B/D column within tile
    const int mb  = (lane >> 4) << 3;     // D row base: 0 or 8

    // ---- Stage 1: hidden (16 x DIN) = xs @ W + bias, node-masked ----------
    for (int nt = wave; nt < DIN / 16; nt += 4) {
        const int n0  = nt * 16;
        const int col = n0 + cl;
        v8f acc = {};
        for (int k0 = 0; k0 < DIN; k0 += 4) {
            v2f a, b;
            a.x = xs[row][k0 + kh];
            a.y = xs[row][k0 + kh + 1];
            b.x = W[(size_t)(k0 + kh) * DIN + col];
            b.y = W[(size_t)(k0 + kh + 1) * DIN + col];
            acc = __builtin_amdgcn_wmma_f32_16x16x4_f32(
                false, a, false, b, (short)0, acc, false, false);
        }
        const float bv = bias[col];
        #pragma unroll
        for (int v = 0; v < 8; ++v) {
            const int m = mb + v;
            hs[m][col] = (acc[v] + bv) * maskf[m];
        }
    }
    __syncthreads();

    // ---- Stage 2: Ldst (16 x DE) = hidden @ We -----------------------------
    for (int nt = wave; nt < DE / 16; nt += 4) {
        const int n0  = nt * 16;
        const int col = n0 + cl;
        v8f acc = {};
        for (int k0 = 0; k0 < DIN; k0 += 4) {
            v2f a, b;
            a.x = hs[row][k0 + kh];
            a.y = hs[row][k0 + kh + 1];
            b.x = We[(size_t)(k0 + kh) * DE + col];
            b.y = We[(size_t)(k0 + kh + 1) * DE + col];
            acc = __builtin_amdgcn_wmma_f32_16x16x4_f32(
                false, a, false, b, (short)0, acc, false, false);
        }
        #pragma unroll
        for (int v = 0; v < 8; ++v) {
            const int m = mb + v;
            Ldst[(size_t)m * DE + col] = acc[v];
        }
    }
    __syncthreads();   // hs is reused by the next side
}

// ---------------------------------------------------------------------------
// Kernel 1: LayerNorm + dual GEMM + edge projections -> Le, Re (B x N x DE)
// Grid: B * (N/16) blocks of 128 threads (4 waves). One 16-node tile/block.
// ---------------------------------------------------------------------------
__global__ void __launch_bounds__(128)
ln_dual_gemm_kernel(const float* __restrict__ X,
                    const uint8_t* __restrict__ node_mask,
                    const float* __restrict__ gamma,
                    const float* __restrict__ beta,
                    const float* __restrict__ Wl, const float* __restrict__ bl,
                    const float* __restrict__ Wr, const float* __restrict__ br,
                    const float* __restrict__ We,   // (2*DIN) x DE
                    float* __restrict__ Le, float* __restrict__ Re)
{
    __shared__ float xs[16][DIN + PAD];
    __shared__ float hs[16][DIN + PAD];
    __shared__ float maskf[16];
    __shared__ float rmu[16];
    __shared__ float rrs[16];

    const int tid  = threadIdx.x;
    const int wave = tid >> 5;
    const int lane = tid & 31;

    const int b  = blockIdx.x / (N_ / 16);
    const int r0 = (blockIdx.x % (N_ / 16)) * 16;

    const float* Xb = X + ((size_t)b * N_ + r0) * DIN;

    // Load 16 x 256 tile into LDS
    for (int i = tid; i < 16 * DIN; i += 128) {
        const int r = i >> 8, c = i & (DIN - 1);
        xs[r][c] = Xb[(size_t)r * DIN + c];
    }
    if (tid < 16)
        maskf[tid] = node_mask[(size_t)b * N_ + r0 + tid] ? 1.0f : 0.0f;
    __syncthreads();

    // Per-row LayerNorm statistics
    if (tid < 16) {
        float s = 0.f, s2 = 0.f;
        for (int c = 0; c < DIN; ++c) {
            const float v = xs[tid][c];
            s += v; s2 += v * v;
        }
        const float mu  = s * (1.0f / DIN);
        const float var = s2 * (1.0f / DIN) - mu * mu;
        rmu[tid] = mu;
        rrs[tid] = rsqrtf(var + LNEPS);
    }
    __syncthreads();

    // Normalize in place: (x - mu) * rsigma * gamma + beta
    for (int i = tid; i < 16 * DIN; i += 128) {
        const int r = i >> 8, c = i & (DIN - 1);
        xs[r][c] = (xs[r][c] - rmu[r]) * rrs[r] * gamma[c] + beta[c];
    }
    __syncthreads();

    float* Lb = Le + ((size_t)b * N_ + r0) * DE;
    float* Rb = Re + ((size_t)b * N_ + r0) * DE;
    gemm_side(xs, hs, Wl, bl, We,                       maskf, Lb, wave, lane);
    gemm_side(xs, hs, Wr, br, We + (size_t)DIN * DE,    maskf, Rb, wave, lane);
}

// ---------------------------------------------------------------------------
// Kernel 2: edge[b,i,j,:] = mask[b,i,j] ? (Le[b,j,:] + Re[b,i,:] + b_edge) : 0
// Pure streaming writer: 604 MB of output, HBM-write bound (~26 us floor).
// One block per (b,i); each 32-lane wave writes one full 512B edge row as
// contiguous non-temporal b128 stores (write-once data must bypass L2 reuse).
// Le rows (393 KB/batch) stay resident in the 192 MB L2 across blocks.
// ---------------------------------------------------------------------------
__global__ void __launch_bounds__(256)
edge_broadcast_kernel(const float* __restrict__ Le,
                      const float* __restrict__ Re,
                      const float* __restrict__ b_edge,
                      const uint8_t* __restrict__ edge_mask,
                      float* __restrict__ out)
{
    __shared__ __align__(16) float r128[DE];

    const int tid = threadIdx.x;
    const int b   = blockIdx.x / N_;
    const int i   = blockIdx.x % N_;

    if (tid < DE)
        r128[tid] = Re[((size_t)b * N_ + i) * DE + tid] + b_edge[tid];
    __syncthreads();

    const float*   Leb = Le + (size_t)b * N_ * DE;
    const uint8_t* em  = edge_mask + ((size_t)b * N_ + i) * N_;
    float*         ob  = out + ((size_t)b * N_ + i) * (size_t)N_ * DE;

    const int jq = tid >> 5;          // which j within a group of 8
    const int q  = (tid & 31) << 2;   // float4 offset within the 128-vector
    const v4f rv = *(const v4f*)&r128[q];

    for (int j0 = 0; j0 < N_; j0 += 8) {
        const int j = j0 + jq;
        const v4f lv = *(const v4f*)(Leb + (size_t)j * DE + q);
        v4f o;
        if (em[j]) {
            o = lv + rv;
        } else {
            o = (v4f){0.f, 0.f, 0.f, 0.f};
        }
        __builtin_nontemporal_store(o, (v4f*)(ob + (size_t)j * DE + q));
    }
}

// ---------------------------------------------------------------------------
extern "C" void kernel_launch(void* const* d_in, const int* in_sizes, int n_in,
                              void* d_out, int out_size, void* d_ws, size_t ws_size,
                              hipStream_t stream)
{
    const float*   X     = (const float*)d_in[0];   // node_features (B,N,DIN)
    const uint8_t* nmask = (const uint8_t*)d_in[1]; // node_mask (B,N) bool
    const uint8_t* emask = (const uint8_t*)d_in[2]; // edge_mask (B,N,N) bool
    const float*   gamma = (const float*)d_in[3];
    const float*   beta  = (const float*)d_in[4];
    const float*   Wl    = (const float*)d_in[5];
    const float*   bl    = (const float*)d_in[6];
    const float*   Wr    = (const float*)d_in[7];
    const float*   br    = (const float*)d_in[8];
    const float*   We    = (const float*)d_in[9];   // (2*DIN, DE)
    const float*   be    = (const float*)d_in[10];

    float* Le = (float*)d_ws;                        // B*N*DE floats
    float* Re = Le + (size_t)B_ * N_ * DE;           // B*N*DE floats

    ln_dual_gemm_kernel<<<B_ * (N_ / 16), 128, 0, stream>>>(
        X, nmask, gamma, beta, Wl, bl, Wr, br, We, Le, Re);

    edge_broadcast_kernel<<<B_ * N_, 256, 0, stream>>>(
        Le, Re, be, emask, (float*)d_out);
}